// SunShapeBlockCodec_15796889714930
// MI455X (gfx1250) — compile-verified
//
#include <hip/hip_runtime.h>

#define N_TOKENS   131072
#define HEAD_DIM   128
#define BLOCK_DIM  8
#define N_BLOCKS   16
#define N_CENT     256
#define N_TILES    16          // 256 centroids / 16 per WMMA tile

typedef __attribute__((ext_vector_type(2))) float v2f;
typedef __attribute__((ext_vector_type(4))) float v4f;
typedef __attribute__((ext_vector_type(8))) float v8f;

// d_ws layout (float units):
//   [0, 32768)      cfrag : centroids in WMMA B-fragment layout (128 KB)
//   [32768, 36864)  csqT  : csqT[b][c&15][t] = ||centroid||^2   (16 KB)
#define WS_CFRAG 0
#define WS_CSQT  32768

// ------------------------------------------------------------------
// Setup: build B-fragment table + transposed centroid norms in d_ws.
// Fragment layout per (b, tile, kbi): 32 lanes x 2 floats (=64 floats),
//   element(lane, e) = C[b][tile*16 + (lane&15)][kbi*4 + (lane>>4)*2 + e]
// matching V_WMMA_F32_16X16X4_F32's B (4x16) operand (lanes 0-15: K0/K1,
// lanes 16-31: K2/K3 across the two VGPRs).
// ------------------------------------------------------------------
__global__ __launch_bounds__(256) void pq_setup(const float* __restrict__ cent,
                                                float* __restrict__ ws) {
  int gid = blockIdx.x * 256 + threadIdx.x;       // 4096 threads: (b, c)
  int b = gid >> 8;
  int c = gid & 255;
  const float* cp = cent + (size_t)(b * N_CENT + c) * BLOCK_DIM;
  v4f lo = *(const v4f*)(cp);
  v4f hi = *(const v4f*)(cp + 4);
  float cs = lo.x*lo.x + lo.y*lo.y + lo.z*lo.z + lo.w*lo.w +
             hi.x*hi.x + hi.y*hi.y + hi.z*hi.z + hi.w*hi.w;
  int t  = c >> 4;
  int cl = c & 15;
  ws[WS_CSQT + (b * 16 + cl) * 16 + t] = cs;
  float v[8] = {lo.x, lo.y, lo.z, lo.w, hi.x, hi.y, hi.z, hi.w};
#pragma unroll
  for (int k = 0; k < 8; ++k) {
    int kbi  = k >> 2;
    int hi2  = (k >> 1) & 1;
    int e    = k & 1;
    int lane = hi2 * 16 + cl;
    ws[WS_CFRAG + (((b * N_TILES + t) * 2 + kbi) << 6) + lane * 2 + e] = v[k];
  }
}

// ------------------------------------------------------------------
// Main: per 16-token tile and per block b, score = ||c||^2 - 2 x.c via
// C-init = ||c||^2 and A pre-scaled by -2; argmin tracked in registers.
// ------------------------------------------------------------------
#define WAVES     4
#define THREADS   (WAVES * 32)      // 128
#define CHUNK_TOK (WAVES * 16)      // 64 tokens per chunk
#define WG_CHUNKS 4
#define WG_TOK    (CHUNK_TOK * WG_CHUNKS)   // 256 tokens per workgroup
#define GRID      (N_TOKENS / WG_TOK)       // 512 workgroups

// LDS (floats): cfrag [0,32768) = 128KB ; xfrag [32768,40960) = 32KB  -> 160KB
#define L_CFRAG 0
#define L_XFRAG 32768

__global__ __launch_bounds__(THREADS) void pq_main(const float* __restrict__ x,
                                                   const int* __restrict__ perm,
                                                   const float* __restrict__ ws,
                                                   int* __restrict__ idx_out) {
  __shared__ __align__(16) float lds[40960];
  const int tid  = threadIdx.x;
  const int lane = tid & 31;
  const int w    = tid >> 5;

  // Stage full centroid fragment table into LDS (32768 floats).
  {
    const v4f* src = (const v4f*)(ws + WS_CFRAG);
    v4f* dst = (v4f*)(lds + L_CFRAG);
    for (int i = tid; i < 8192; i += THREADS) dst[i] = src[i];
  }

  // Loader mapping: this thread gathers permuted column i = tid for all
  // tokens of a chunk, writing straight into A-fragment layout.
  const int p_i    = perm[tid & 127];
  const int li_b   = (tid & 127) >> 3;
  const int li_k   = tid & 7;
  const int li_kbi = li_k >> 2;
  const int li_hi  = (li_k >> 1) & 1;
  const int li_e   = li_k & 1;

  const int wg_base = blockIdx.x * WG_TOK;
  const int cl = lane & 15;

  for (int chunk = 0; chunk < WG_CHUNKS; ++chunk) {
    const int n0 = wg_base + chunk * CHUNK_TOK;
    __syncthreads();   // previous chunk's readers done before overwrite
    for (int m = 0; m < CHUNK_TOK; ++m) {
      float xv = -2.0f * x[(size_t)(n0 + m) * HEAD_DIM + p_i];
      int wv = m >> 4;
      int mm = m & 15;
      int lanew = li_hi * 16 + mm;
      lds[L_XFRAG + (((wv * N_BLOCKS + li_b) * 2 + li_kbi) << 6) + lanew * 2 + li_e] = xv;
    }
    __syncthreads();

    const int base_n = n0 + w * 16;
#pragma unroll 1
    for (int b = 0; b < N_BLOCKS; ++b) {
      // A fragments (K=0..3 and K=4..7) for this wave's 16 tokens.
      const v2f* xa = (const v2f*)(lds + L_XFRAG + (((w * N_BLOCKS + b) * 2) << 6));
      v2f a0 = xa[lane];
      v2f a1 = xa[32 + lane];
      const float* csr = ws + WS_CSQT + (b * 16 + cl) * 16;

      float minval[8];
      int   minidx[8];
#pragma unroll
      for (int r = 0; r < 8; ++r) { minval[r] = __builtin_inff(); minidx[r] = 0; }

      // 4 groups of 4 tiles: one v4f of ||c||^2 per group (prefetched one
      // group ahead to hide L2 latency), 4 independent WMMA accumulation
      // chains in flight, bounded live compare results.
      v4f cs = *(const v4f*)(csr);
#pragma unroll 1
      for (int tt = 0; tt < 4; ++tt) {
        // Prefetch next group's ||c||^2 (wraps to group 0 on last trip).
        v4f csn = *(const v4f*)(csr + (((tt + 1) & 3) << 2));
        float cqs[4] = {cs.x, cs.y, cs.z, cs.w};
#pragma unroll
        for (int u = 0; u < 4; ++u) {
          int t = tt * 4 + u;
          const v2f* cf = (const v2f*)(lds + L_CFRAG + (((b * N_TILES + t) * 2) << 6));
          v2f b0 = cf[lane];
          v2f b1 = cf[32 + lane];
          float cq = cqs[u];
          v8f acc = {cq, cq, cq, cq, cq, cq, cq, cq};
          acc = __builtin_amdgcn_wmma_f32_16x16x4_f32(false, a0, false, b0,
                                                      (short)0, acc, false, false);
          acc = __builtin_amdgcn_wmma_f32_16x16x4_f32(false, a1, false, b1,
                                                      (short)0, acc, false, false);
          int idxc = t * 16 + cl;
#pragma unroll
          for (int r = 0; r < 8; ++r) {
            bool lt = acc[r] < minval[r];
            minval[r] = lt ? acc[r] : minval[r];
            minidx[r] = lt ? idxc   : minidx[r];
          }
        }
        cs = csn;
      }

      // Cross-lane argmin over the 16 lanes sharing each row group,
      // lexicographic tie-break on index (matches jnp.argmin first-min).
#pragma unroll
      for (int s = 1; s < 16; s <<= 1) {
#pragma unroll
        for (int r = 0; r < 8; ++r) {
          float ov = __shfl_xor(minval[r], s, 32);
          int   oi = __shfl_xor(minidx[r], s, 32);
          bool take = (ov < minval[r]) || (ov == minval[r] && oi < minidx[r]);
          minval[r] = take ? ov : minval[r];
          minidx[r] = take ? oi : minidx[r];
        }
      }

      // Lane (lane&15)==r writes row r of its half (rows 0-7 low, 8-15 high).
      int r = lane & 7;
      int mi = minidx[0];
#pragma unroll
      for (int q = 1; q < 8; ++q) mi = (r == q) ? minidx[q] : mi;
      if ((lane & 15) < 8) {
        int row = (lane >> 4) * 8 + r;
        idx_out[(size_t)(base_n + row) * N_BLOCKS + b] = mi;
      }
    }
  }
}

// ------------------------------------------------------------------
// Recon: one thread per (token, block); gather codeword, scatter via perm
// (recon[:, perm[i]] = recon_p[:, i]  <=>  recon[:, :] = recon_p[:, inv_perm]).
// ------------------------------------------------------------------
__global__ __launch_bounds__(256) void pq_recon(const float* __restrict__ cent,
                                                const int* __restrict__ perm,
                                                const int* __restrict__ idx_in,
                                                float* __restrict__ recon) {
  size_t gid = (size_t)blockIdx.x * 256 + threadIdx.x;   // 2M threads
  int b = (int)(gid & 15);
  size_t n = gid >> 4;
  int idx = idx_in[gid];
  const float* cp = cent + ((size_t)(b * N_CENT + idx)) * BLOCK_DIM;
  v4f lo = *(const v4f*)cp;
  v4f hi = *(const v4f*)(cp + 4);
  float v[8] = {lo.x, lo.y, lo.z, lo.w, hi.x, hi.y, hi.z, hi.w};
  const int* pp = perm + b * BLOCK_DIM;
  float* out = recon + n * HEAD_DIM;
#pragma unroll
  for (int d = 0; d < 8; ++d) out[pp[d]] = v[d];
}

// ------------------------------------------------------------------
extern "C" void kernel_launch(void* const* d_in, const int* in_sizes, int n_in,
                              void* d_out, int out_size, void* d_ws, size_t ws_size,
                              hipStream_t stream) {
  (void)in_sizes; (void)n_in; (void)out_size; (void)ws_size;
  const float* x    = (const float*)d_in[0];
  const float* cent = (const float*)d_in[1];
  const int*   perm = (const int*)d_in[2];
  float* ws      = (float*)d_ws;
  int*   idx_out = (int*)d_out;
  float* recon   = (float*)d_out + (size_t)N_TOKENS * N_BLOCKS;

  pq_setup<<<16, 256, 0, stream>>>(cent, ws);
  pq_main<<<GRID, THREADS, 0, stream>>>(x, perm, ws, idx_out);
  pq_recon<<<(N_TOKENS * N_BLOCKS) / 256, 256, 0, stream>>>(cent, perm, idx_out, recon);
}